// FirstOrderDCN_13829794693639
// MI455X (gfx1250) — compile-verified
//
#include <hip/hip_runtime.h>
#include <hip/hip_bf16.h>
#include <math.h>

// ---------------------------------------------------------------------------
// FirstOrderDCN for MI455X (gfx1250, wave32, WMMA).
//  conv1..conv4 : implicit-GEMM 3x3 SAME conv via V_WMMA_F32_16X16X4_F32,
//                 K ordered as (tap, ic) with weights pre-packed to
//                 wp[ocPad][tap][ic] so the hot loop has no divisions, no
//                 divergent branches, pure pointer-increment addressing,
//                 and a b64 weight load per wmma.
//  conv4 epilogue fuses 10*tanh + flip(offsets) add and sigmoid(mask).
//  dcn          : modulated deformable conv, LDS-staged group weights.
// ---------------------------------------------------------------------------

typedef __attribute__((ext_vector_type(2))) float v2f;
typedef __attribute__((ext_vector_type(8))) float v8f;

#define BB 2
#define HH 128
#define WW 128
#define OG 8
#define HW (HH * WW)

// Repack w[oc][ic][3][3] -> wp[ocPad][tap][ic], zero-fill oc >= Cout.
__global__ __launch_bounds__(256) void repack_w_kernel(
    const float* __restrict__ w, float* __restrict__ wp,
    int Cin, int Cout, int CoutPad)
{
    const int i = blockIdx.x * blockDim.x + threadIdx.x;
    const int total = CoutPad * 9 * Cin;
    if (i >= total) return;
    const int ic  = i % Cin;
    const int r   = i / Cin;
    const int tap = r % 9;
    const int oc  = r / 9;
    wp[i] = (oc < Cout) ? w[((size_t)oc * Cin + ic) * 9 + tap] : 0.f;
}

__device__ __forceinline__ v8f wmma_f32(v2f a, v2f b, v8f c) {
    return __builtin_amdgcn_wmma_f32_16x16x4_f32(false, a, false, b,
                                                 (short)0, c, false, false);
}

// mode: 1 = leaky_relu(0.1) -> out ;
//       2 = conv4 tail: ch<144 -> dcnoff = 10*tanh(v)+offsets[143-ch]
//                       ch>=144 -> maskout = sigmoid(v)
__global__ __launch_bounds__(256) void conv3x3_wmma_kernel(
    const float* __restrict__ srcA, int CinA,
    const float* __restrict__ srcB, int CinB,
    const float* __restrict__ wp,       // packed [CoutPad][9][Cin]
    const float* __restrict__ bias,     // [Cout]
    int Cout, int mode,
    float* __restrict__ out,            // [B][Cout][H][W]   (mode 1)
    const float* __restrict__ offsets,  // [B][144][H][W]    (mode 2)
    float* __restrict__ dcnoff,         // [B][144][H][W]    (mode 2)
    float* __restrict__ maskout)        // [B][72][H][W]     (mode 2)
{
    const int Cin    = CinA + CinB;
    const int tilesN = (Cout + 15) >> 4;

    int wid = (blockIdx.x * blockDim.x + threadIdx.x) >> 5;
    const int lane = threadIdx.x & 31;
    const int nT = wid % tilesN;    wid /= tilesN;
    const int xT = wid % (WW / 16); wid /= (WW / 16);
    const int y  = wid % HH;        wid /= HH;
    const int b  = wid;
    if (b >= BB) return;            // wave-uniform; grids launched exact

    const int m16 = lane & 15;      // A: M row (x pos) | B: N col (oc)
    const int h   = lane >> 4;      // lane half -> K half
    const int x0  = xT * 16;
    const int oc  = nT * 16 + m16;

    // Lane's packed-weight base: k = tap*Cin + ic; this lane covers
    // K = k0+2h+v, v=0,1 -> two consecutive ic -> contiguous b64 load.
    const float* bk = wp + (size_t)oc * 9 * Cin + 2 * h;

    v8f acc = {};

#pragma unroll
    for (int tap = 0; tap < 9; ++tap) {
        const int ky = tap / 3;            // compile-time (tap loop unrolled)
        const int kx = tap - ky * 3;
        const int yy = y + ky - 1;
        const int xx = x0 + m16 + kx - 1;
        const float vf =
            ((unsigned)yy < (unsigned)HH && (unsigned)xx < (unsigned)WW) ? 1.f : 0.f;
        const int cyy = min(max(yy, 0), HH - 1);
        const int cxx = min(max(xx, 0), WW - 1);

        // --- source A channels: ic in [0, CinA) ---
        {
            const float* pa =
                srcA + (((size_t)b * CinA + 2 * h) * HH + cyy) * WW + cxx;
            const int iters = CinA >> 2;
#pragma unroll 4
            for (int i = 0; i < iters; ++i) {
                v2f a, bf;
                a[0] = pa[0] * vf;
                a[1] = pa[HW] * vf;
                bf = *reinterpret_cast<const v2f*>(bk);
                acc = wmma_f32(a, bf, acc);
                pa += (size_t)4 * HW;
                bk += 4;
            }
        }
        // --- source B channels: ic in [CinA, Cin) ---
        if (CinB > 0) {
            const float* pa =
                srcB + (((size_t)b * CinB + 2 * h) * HH + cyy) * WW + cxx;
            const int iters = CinB >> 2;
#pragma unroll 4
            for (int i = 0; i < iters; ++i) {
                v2f a, bf;
                a[0] = pa[0] * vf;
                a[1] = pa[HW] * vf;
                bf = *reinterpret_cast<const v2f*>(bk);
                acc = wmma_f32(a, bf, acc);
                pa += (size_t)4 * HW;
                bk += 4;
            }
        }
    }

    // C/D layout: lane -> N (oc), VGPR v -> M = v + 8*h (spatial x offset)
    if (oc < Cout) {
        const float bv = bias[oc];
#pragma unroll
        for (int v = 0; v < 8; ++v) {
            const int xm = x0 + v + 8 * h;
            float val = acc[v] + bv;
            if (mode == 1) {
                val = val > 0.f ? val : 0.1f * val;
                out[(((size_t)b * Cout + oc) * HH + y) * WW + xm] = val;
            } else {
                if (oc < 144) {
                    const float flip =
                        offsets[(((size_t)b * 144 + (143 - oc)) * HH + y) * WW + xm];
                    dcnoff[(((size_t)b * 144 + oc) * HH + y) * WW + xm] =
                        10.f * tanhf(val) + flip;
                } else {
                    maskout[(((size_t)b * 72 + (oc - 144)) * HH + y) * WW + xm] =
                        1.f / (1.f + expf(-val));
                }
            }
        }
    }
}

// Modulated deformable conv: thread = (b, g, y, x), produces 8 out channels.
__global__ __launch_bounds__(256) void dcn_kernel(
    const float* __restrict__ x,       // [B,64,H,W]
    const float* __restrict__ dcnoff,  // [B,144,H,W]  layout (g,k,(y,x))
    const float* __restrict__ msk,     // [B,72,H,W]   layout (g,k)
    const float* __restrict__ wd,      // [64][8][3][3]
    const float* __restrict__ bd,      // [64]
    float* __restrict__ out)           // [B,64,H,W]
{
    __shared__ float sw[8 * 72];       // this block's group weights [o][c*9+k]

    const int idx = blockIdx.x * blockDim.x + threadIdx.x;
    const int xx = idx & (WW - 1);
    int rest = idx >> 7;
    const int y = rest & (HH - 1);
    rest >>= 7;
    const int g = rest & (OG - 1);
    const int b = rest >> 3;

    // g is uniform across the block (H*W per group >> blockDim)
    for (int i = threadIdx.x; i < 8 * 72; i += blockDim.x)
        sw[i] = wd[(size_t)(g * 8) * 72 + i];
    __syncthreads();

    float acc[8] = {0.f, 0.f, 0.f, 0.f, 0.f, 0.f, 0.f, 0.f};
    const float* xg = x + ((size_t)b * 64 + g * 8) * (size_t)HW;

#pragma unroll
    for (int k = 0; k < 9; ++k) {
        const size_t obase = (((size_t)b * 144 + (g * 9 + k) * 2) * HH + y) * WW + xx;
        const float oy = dcnoff[obase];
        const float ox = dcnoff[obase + (size_t)HW];
        const float m  = msk[(((size_t)b * 72 + g * 9 + k) * HH + y) * WW + xx];

        const float py = oy + (float)(k / 3 - 1) + (float)y;
        const float px = ox + (float)(k % 3 - 1) + (float)xx;
        const float fy = floorf(py), fx = floorf(px);
        const int iy0 = (int)fy, ix0 = (int)fx;
        const float ly = py - fy, lx = px - fx;

        const float w00 = (1.f - ly) * (1.f - lx) * m;
        const float w01 = (1.f - ly) * lx * m;
        const float w10 = ly * (1.f - lx) * m;
        const float w11 = ly * lx * m;

        const bool vy0 = (iy0 >= 0) && (iy0 < HH);
        const bool vy1 = (iy0 + 1 >= 0) && (iy0 + 1 < HH);
        const bool vx0 = (ix0 >= 0) && (ix0 < WW);
        const bool vx1 = (ix0 + 1 >= 0) && (ix0 + 1 < WW);
        const int cy0 = min(max(iy0, 0), HH - 1);
        const int cy1 = min(max(iy0 + 1, 0), HH - 1);
        const int cx0 = min(max(ix0, 0), WW - 1);
        const int cx1 = min(max(ix0 + 1, 0), WW - 1);

#pragma unroll
        for (int c = 0; c < 8; ++c) {
            const float* xc = xg + (size_t)c * HW;
            float s = 0.f;
            if (vy0) {
                const float* row = xc + (size_t)cy0 * WW;
                if (vx0) s += w00 * row[cx0];
                if (vx1) s += w01 * row[cx1];
            }
            if (vy1) {
                const float* row = xc + (size_t)cy1 * WW;
                if (vx0) s += w10 * row[cx0];
                if (vx1) s += w11 * row[cx1];
            }
#pragma unroll
            for (int o = 0; o < 8; ++o)
                acc[o] += s * sw[o * 72 + c * 9 + k];
        }
    }

    const size_t ob = (((size_t)b * 64 + g * 8) * HH + y) * WW + xx;
#pragma unroll
    for (int o = 0; o < 8; ++o)
        out[ob + (size_t)o * HW] = acc[o] + bd[g * 8 + o];
}

extern "C" void kernel_launch(void* const* d_in, const int* in_sizes, int n_in,
                              void* d_out, int out_size, void* d_ws, size_t ws_size,
                              hipStream_t stream) {
    const float* x   = (const float*)d_in[0];   // [2,64,128,128]
    const float* ef  = (const float*)d_in[1];   // [2,128,128,128]
    const float* off = (const float*)d_in[2];   // [2,144,128,128]
    const float* w1  = (const float*)d_in[3];
    const float* b1  = (const float*)d_in[4];
    const float* w2  = (const float*)d_in[5];
    const float* b2  = (const float*)d_in[6];
    const float* w3  = (const float*)d_in[7];
    const float* b3  = (const float*)d_in[8];
    const float* w4  = (const float*)d_in[9];
    const float* b4  = (const float*)d_in[10];
    const float* wd  = (const float*)d_in[11];
    const float* bd  = (const float*)d_in[12];
    float* out = (float*)d_out;

    const size_t plane = (size_t)BB * HW;       // 32768
    float* t1  = (float*)d_ws;                  // [2,64,H,W]   8 MB
    float* t2  = t1 + plane * 64;               // [2,64,H,W]   8 MB
    float* dcn = t2 + plane * 64;               // [2,144,H,W] 18 MB
    float* msk = dcn + plane * 144;             // [2,72,H,W]   9 MB
    float* wp1 = msk + plane * 72;              // 64*9*272
    float* wp2 = wp1 + (size_t)64 * 9 * 272;    // 64*9*64
    float* wp3 = wp2 + (size_t)64 * 9 * 64;     // 64*9*64
    float* wp4 = wp3 + (size_t)64 * 9 * 64;     // 224*9*64

    const dim3 blk(256);

    // --- weight repacks (ordered on stream before their convs) ---
    {
        int t = 64 * 9 * 272;
        repack_w_kernel<<<(t + 255) / 256, blk, 0, stream>>>(w1, wp1, 272, 64, 64);
        t = 64 * 9 * 64;
        repack_w_kernel<<<(t + 255) / 256, blk, 0, stream>>>(w2, wp2, 64, 64, 64);
        repack_w_kernel<<<(t + 255) / 256, blk, 0, stream>>>(w3, wp3, 64, 64, 64);
        t = 224 * 9 * 64;
        repack_w_kernel<<<(t + 255) / 256, blk, 0, stream>>>(w4, wp4, 64, 216, 224);
    }

    // waves = B * H * (W/16) * (CoutPad/16); all exact multiples of 8 waves.
    const int waves64  = BB * HH * (WW / 16) * 4;   // 8192  -> 1024 blocks
    const int waves216 = BB * HH * (WW / 16) * 14;  // 28672 -> 3584 blocks

    conv3x3_wmma_kernel<<<waves64 * 32 / 256, blk, 0, stream>>>(
        ef, 2 * 64, off, 144, wp1, b1, 64, /*mode=*/1, t1, nullptr, nullptr, nullptr);

    conv3x3_wmma_kernel<<<waves64 * 32 / 256, blk, 0, stream>>>(
        t1, 64, nullptr, 0, wp2, b2, 64, /*mode=*/1, t2, nullptr, nullptr, nullptr);

    conv3x3_wmma_kernel<<<waves64 * 32 / 256, blk, 0, stream>>>(
        t2, 64, nullptr, 0, wp3, b3, 64, /*mode=*/1, t1, nullptr, nullptr, nullptr);

    conv3x3_wmma_kernel<<<waves216 * 32 / 256, blk, 0, stream>>>(
        t1, 64, nullptr, 0, wp4, b4, 216, /*mode=*/2, nullptr, off, dcn, msk);

    dcn_kernel<<<(BB * OG * HH * WW) / 256, blk, 0, stream>>>(
        x, dcn, msk, wd, bd, out);
}